// MyCS_fc_28020366639787
// MI455X (gfx1250) — compile-verified
//
#include <hip/hip_runtime.h>
#include <hip/hip_bf16.h>
#include <math.h>

// Problem constants (match reference)
static constexpr int kC = 4096;        // channels
static constexpr int kN = 16384;       // batch
static constexpr int kSlices = 64;     // partial-sum slices over N
static constexpr int kRowsPerSlice = kN / kSlices;  // 256

typedef float v2f __attribute__((ext_vector_type(2)));
typedef float v8f __attribute__((ext_vector_type(8)));

// ---------------------------------------------------------------------------
// Kernel 1: partial column sums. grid = (kC/256, kSlices), block = 256.
// Each thread owns one column within a slice of 256 rows; per-row the block
// reads 1024 contiguous bytes (fully coalesced b32 stream of x: 256 MB).
// ---------------------------------------------------------------------------
__global__ void __launch_bounds__(256) colsum_partial(const float* __restrict__ x,
                                                      float* __restrict__ part) {
    const int c     = blockIdx.x * 256 + threadIdx.x;   // column
    const int slice = blockIdx.y;
    const float* xp = x + (size_t)slice * kRowsPerSlice * kC + c;
    float s = 0.0f;
#pragma unroll 8
    for (int r = 0; r < kRowsPerSlice; ++r) {
        s += xp[(size_t)r * kC];
    }
    part[slice * kC + c] = s;
}

// ---------------------------------------------------------------------------
// Kernel 2: finalize g[c] = (1/N) * sum_slices part[s][c]. grid = kC/256.
// ---------------------------------------------------------------------------
__global__ void __launch_bounds__(256) colsum_final(const float* __restrict__ part,
                                                    float* __restrict__ g) {
    const int c = blockIdx.x * 256 + threadIdx.x;
    float s = 0.0f;
#pragma unroll 8
    for (int i = 0; i < kSlices; ++i) {
        s += part[i * kC + c];
    }
    g[c] = s * (1.0f / (float)kN);
}

// ---------------------------------------------------------------------------
// Kernel 3: y = W @ g + b ; s = sigmoid(scale * y) via V_WMMA_F32_16X16X4_F32.
//
// Per wave: 16 output rows (m0..m0+15). A = W tile (16x4), documented layout:
//   lane L: M = L&15, VGPR v holds K = v + 2*(L>>4)  -> float2 at W[row][k+2*half].
// B = g broadcast into all 16 columns (4x16). Column mapping is irrelevant
// (all columns equal); K mapping follows the same half-wave K split:
//   lane L: VGPR v holds K = v + 2*(L>>4)            -> float2 at g[k+2*half].
// D (16x16 f32, documented): lane L holds N = L&15, M = v + 8*(L>>4); every
// column is identical, so lanes with (L&15)==0 emit the 16 y values.
// grid = kC/128 = 32 blocks of 8 waves -> 256 waves, W read exactly once.
// ---------------------------------------------------------------------------
__global__ void __launch_bounds__(256) matvec_sigmoid(const float* __restrict__ W,
                                                      const float* __restrict__ g,
                                                      const float* __restrict__ bias,
                                                      const int* __restrict__ sf,
                                                      float* __restrict__ s_ws,
                                                      float* __restrict__ s_out) {
    __shared__ float gs[kC];
    const int tid = threadIdx.x;
    for (int i = tid; i < kC; i += 256) gs[i] = g[i];
    __syncthreads();

    const int lane = tid & 31;
    const int wave = tid >> 5;
    const int half = lane >> 4;       // 0: K={0,1}, 1: K={2,3}
    const int mrow = lane & 15;       // M within tile
    const int m0   = (blockIdx.x * 8 + wave) * 16;

    const float* Wr = W + (size_t)(m0 + mrow) * kC + 2 * half;
    const float* gp = gs + 2 * half;

    v8f acc = {};
    for (int k = 0; k < kC; k += 4) {
        v2f a = *(const v2f*)(Wr + k);   // global b64, 16 rows striped over lanes
        v2f b = *(const v2f*)(gp + k);   // LDS b64 broadcast (no bank conflict)
        acc = __builtin_amdgcn_wmma_f32_16x16x4_f32(
            /*neg_a=*/false, a, /*neg_b=*/false, b,
            /*c_mod=*/(short)0, acc, /*reuse_a=*/false, /*reuse_b=*/false);
    }

    if (mrow == 0) {                  // lane 0 -> M 0..7, lane 16 -> M 8..15
        const float sfv = (float)sf[0];
        const int base = m0 + 8 * half;
#pragma unroll
        for (int v = 0; v < 8; ++v) {
            float y  = acc[v] + bias[base + v];
            float sv = 1.0f / (1.0f + expf(-sfv * y));
            s_ws[base + v]  = sv;     // for kernel 4
            s_out[base + v] = sv;     // x_scale output (tail of d_out)
        }
    }
}

// ---------------------------------------------------------------------------
// Kernel 4: out[r][c] = x[r][c] * s[c], float4-vectorized (b128 stream).
// grid = N*C/4/256 = 65536 blocks. s (16 KB) stays resident in L2.
// ---------------------------------------------------------------------------
__global__ void __launch_bounds__(256) scale_kernel(const float* __restrict__ x,
                                                    const float* __restrict__ s,
                                                    float* __restrict__ out) {
    const size_t i = ((size_t)blockIdx.x * 256 + threadIdx.x) * 4;
    const int c = (int)(i & (size_t)(kC - 1));
    float4 xv = *(const float4*)(x + i);
    float4 sv = *(const float4*)(s + c);
    float4 ov;
    ov.x = xv.x * sv.x;
    ov.y = xv.y * sv.y;
    ov.z = xv.z * sv.z;
    ov.w = xv.w * sv.w;
    *(float4*)(out + i) = ov;
}

// ---------------------------------------------------------------------------
// Launch: 4 kernels in stream order (stream order = dependency order).
// Workspace layout (floats): [0, 64*C) partials | [64*C, 65*C) g | [65*C, 66*C) s
// Total ws use: 66*4096*4 B ~= 1.06 MB. All outputs fully rewritten per call.
// ---------------------------------------------------------------------------
extern "C" void kernel_launch(void* const* d_in, const int* in_sizes, int n_in,
                              void* d_out, int out_size, void* d_ws, size_t ws_size,
                              hipStream_t stream) {
    const float* x    = (const float*)d_in[0];   // [N, C]
    const float* W    = (const float*)d_in[1];   // [C, C]
    const float* bias = (const float*)d_in[2];   // [C]
    const int*   sf   = (const int*)d_in[3];     // scalar scale_factor

    float* out      = (float*)d_out;             // [N*C] x_out, then [C] x_scale
    float* out_tail = out + (size_t)kN * kC;

    float* ws   = (float*)d_ws;
    float* part = ws;                // 64*C
    float* g    = ws + 64 * kC;      // C
    float* s    = ws + 65 * kC;      // C

    colsum_partial<<<dim3(kC / 256, kSlices), 256, 0, stream>>>(x, part);
    colsum_final<<<kC / 256, 256, 0, stream>>>(part, g);
    matvec_sigmoid<<<kC / 128, 256, 0, stream>>>(W, g, bias, sf, s, out_tail);
    scale_kernel<<<((size_t)kN * kC) / 4 / 256, 256, 0, stream>>>(x, s, out);
}